// SimplifiedL2Adapter_1795296329755
// MI455X (gfx1250) — compile-verified
//
#include <hip/hip_runtime.h>
#include <hip/hip_bf16.h>
#include <math.h>

typedef __attribute__((ext_vector_type(2))) float v2f;
typedef __attribute__((ext_vector_type(4))) float v4f;
typedef __attribute__((ext_vector_type(8))) float v8f;

#define BATCH 4
#define SEQ   8192
#define DIM   768
#define MEMSZ 512

// ---------------------------------------------------------------------------
// Kernel 1: importance[s] = (1/B) * sum_b ||hidden[b,s,:]||_2
//
// One wave handles 16 tokens. Sum-of-squares over D is computed on the tensor
// cores: V_WMMA_F32_16X16X4_F32 with A == B (register layouts of A[16x4] and
// B[4x16] are exact transposes of each other), so D = A * A^T and the diagonal
// D[m][m] accumulates sum_k A[m][k]^2. Two accumulators break the WMMA->WMMA
// accumulation chain. Diagonal of the 16x16 f32 C-tile lives at
// (vgpr = m&7, lane = m<8 ? m : m+16), i.e. per-lane element c[lane&7] on
// lanes 0-7 and 24-31; a ds_bpermute moves it to lane m.
// ---------------------------------------------------------------------------
__global__ __launch_bounds__(128) void importance_kernel(
    const float* __restrict__ hidden, float* __restrict__ importance) {
  const int lane = threadIdx.x & 31;
  const int wave = threadIdx.x >> 5;
  const int t0   = (blockIdx.x * 4 + wave) * 16;   // first token of this wave's tile
  const int m    = lane & 15;                      // token within tile
  const int half = lane >> 4;                      // 0: K0/K1, 1: K2/K3

  float acc = 0.0f;
  for (int b = 0; b < BATCH; ++b) {
    const float* row = hidden + ((size_t)b * SEQ + (size_t)(t0 + m)) * DIM + 2 * half;
    v8f c0 = {};
    v8f c1 = {};
#pragma unroll 4
    for (int c = 0; c < DIM / 4; c += 2) {
      v2f a0 = *(const v2f*)(row + 4 * c);         // K = 4c+2*half, +1   (b64 load)
      v2f a1 = *(const v2f*)(row + 4 * (c + 1));
      // D = A * A^T + C  (same regs as A and B -> B is A transposed)
      c0 = __builtin_amdgcn_wmma_f32_16x16x4_f32(false, a0, false, a0,
                                                 (short)0, c0, false, false);
      c1 = __builtin_amdgcn_wmma_f32_16x16x4_f32(false, a1, false, a1,
                                                 (short)0, c1, false, false);
    }
    // Per-lane pick of the diagonal candidate held by this lane.
    float dval = c0[lane & 7] + c1[lane & 7];
    // Lane l (l<16) fetches diag(token l): src lane = l (l<8) else l+16.
    int src = (lane < 8) ? lane : (lane + 16);
    int di  = __builtin_amdgcn_ds_bpermute(src << 2, __builtin_bit_cast(int, dval));
    float diag = __builtin_bit_cast(float, di);
    acc += sqrtf(diag);
  }
  if (lane < 16) importance[t0 + lane] = acc * 0.25f;
}

// ---------------------------------------------------------------------------
// Kernel 2: exact top-k by ranking. rank(s) = #{ j : imp[j] > imp[s]  or
// (imp[j] == imp[s] and j < s) }. Ranks are a permutation (ties broken by
// index), so each rank < K is written by exactly one thread — reproduces
// jax.lax.top_k ordering exactly. importance (32 KB) cached in LDS.
// ---------------------------------------------------------------------------
__global__ __launch_bounds__(256) void topk_rank_kernel(
    const float* __restrict__ importance, int* __restrict__ topk_idx) {
  __shared__ float simp[SEQ];
  const int tid = threadIdx.x;
  for (int i = tid; i < SEQ; i += 256) simp[i] = importance[i];
  __syncthreads();

  const int s = blockIdx.x * 256 + tid;
  const float v = simp[s];
  int rank = 0;
#pragma unroll 8
  for (int j = 0; j < SEQ; ++j) {
    float u = simp[j];
    rank += (u > v) || (u == v && j < s);
  }
  if (rank < MEMSZ) topk_idx[rank] = s;
}

// ---------------------------------------------------------------------------
// Kernel 3: out[r,:] = (1/B) * sum_b hidden[b, topk_idx[r], :]
// 512 blocks (one per output row) x 192 threads (one float4 each).
// Rows were just streamed in kernel 1 -> hidden (96 MB) largely resident in
// the 192 MB L2, so this 6 MB re-read is cheap.
// ---------------------------------------------------------------------------
__global__ __launch_bounds__(192) void gather_mean_kernel(
    const float* __restrict__ hidden, const int* __restrict__ topk_idx,
    float* __restrict__ out) {
  const int r = blockIdx.x;       // 0..511
  const int q = threadIdx.x;      // 0..191 float4 lanes
  const int tok = topk_idx[r];
  const size_t rowOff = (size_t)tok * DIM;

  v4f a = {};
#pragma unroll
  for (int b = 0; b < BATCH; ++b) {
    const v4f* p = (const v4f*)(hidden + (size_t)b * SEQ * DIM + rowOff);
    a += p[q];
  }
  ((v4f*)out)[(size_t)r * (DIM / 4) + q] = a * 0.25f;
}

// ---------------------------------------------------------------------------
extern "C" void kernel_launch(void* const* d_in, const int* in_sizes, int n_in,
                              void* d_out, int out_size, void* d_ws, size_t ws_size,
                              hipStream_t stream) {
  (void)in_sizes; (void)n_in; (void)out_size; (void)ws_size;
  const float* hidden = (const float*)d_in[0];
  // d_in[1] (memory) is dead: k == MEMORY_SIZE, every row is overwritten.
  float* out = (float*)d_out;

  float* importance = (float*)d_ws;                          // 8192 floats
  int*   topk_idx   = (int*)((char*)d_ws + SEQ * sizeof(float)); // 512 ints

  importance_kernel<<<SEQ / 64, 128, 0, stream>>>(hidden, importance);
  topk_rank_kernel<<<SEQ / 256, 256, 0, stream>>>(importance, topk_idx);
  gather_mean_kernel<<<MEMSZ, DIM / 4, 0, stream>>>(hidden, topk_idx, out);
}